// MipRayMarcher2_54906861912227
// MI455X (gfx1250) — compile-verified
//
#include <hip/hip_runtime.h>
#include <cstdint>
#include <cstddef>

// Problem constants (from reference): B=4, R=16384, S=48, CSEM=4
#define EPS_F 1e-6f
constexpr int B_ = 4;
constexpr int R_ = 16384;
constexpr int S_ = 48;
constexpr int NRAYS = B_ * R_;   // 65536
constexpr int NM1 = S_ - 1;      // 47 intervals
constexpr int RPB = 8;           // rays (waves) per 256-thread block
static_assert(NRAYS % RPB == 0, "");

// ---------------------------------------------------------------------------
// Kernel 1: init workspace min/max (ints holding float bit patterns;
// all depths are > 0 so IEEE bit order == signed-int order).
// ---------------------------------------------------------------------------
__global__ void k_init_minmax(int* ws) {
  ws[0] = 0x7F800000;  // +inf  (running min)
  ws[1] = 0x00000000;  // 0.0f  (running max; all depths >= 0.5)
}

// ---------------------------------------------------------------------------
// Kernel 2: global min/max over the depths tensor (B*R*S floats).
// ---------------------------------------------------------------------------
__global__ __launch_bounds__(256)
void k_depth_minmax(const float* __restrict__ depths, int* __restrict__ ws, int n) {
  int tid = blockIdx.x * blockDim.x + threadIdx.x;
  int stride = gridDim.x * blockDim.x;
  float mn = 3.0e38f, mx = -3.0e38f;
  for (int i = tid; i < n; i += stride) {
    float v = depths[i];
    mn = fminf(mn, v);
    mx = fmaxf(mx, v);
  }
  for (int off = 16; off; off >>= 1) {
    mn = fminf(mn, __shfl_xor(mn, off));
    mx = fmaxf(mx, __shfl_xor(mx, off));
  }
  if ((threadIdx.x & 31) == 0) {
    atomicMin(&ws[0], __float_as_int(mn));
    atomicMax(&ws[1], __float_as_int(mx));
  }
}

// ---------------------------------------------------------------------------
// CDNA5 async global -> LDS copy (ASYNCcnt path).
// lds_off: byte address in LDS (low 32 bits of the generic pointer ==
// offset relative to the workgroup's LDS base, per the flat-aperture rule).
// ---------------------------------------------------------------------------
__device__ __forceinline__ uint32_t lds_addr32(const void* p) {
  return (uint32_t)(uintptr_t)p;
}
__device__ __forceinline__ void async_copy_b128(uint32_t lds_off, const void* gsrc) {
  asm volatile("global_load_async_to_lds_b128 %0, %1, off"
               :: "v"(lds_off), "v"(gsrc)
               : "memory");
}
__device__ __forceinline__ void wait_asynccnt0() {
  asm volatile("s_wait_asynccnt 0x0" ::: "memory");
}

// ---------------------------------------------------------------------------
// Main kernel: one wave32 per ray.
//   chunk A: lane holds sample s = lane        (s = 0..31), interval i = lane
//   chunk B: lane holds sample s = 32 + lane   (lanes 0..15), interval 32+lane
// ---------------------------------------------------------------------------
__device__ __forceinline__ void compute_interval(
    int lv,
    float c0, float c0n, float c1, float c1n, float c2, float c2n,
    float d,  float dn,  float z,  float zn,
    const float4& s, const float4& sn,
    bool active,
    float& mc0, float& mc1, float& mc2, float& mz,
    float& ms0, float& ms1, float& ms2, float& ms3,
    float& alpha)
{
  mc0 = (c0 + c0n) * 0.5f;
  mc1 = (c1 + c1n) * 0.5f;
  mc2 = (c2 + c2n) * 0.5f;
  float md = (d + dn) * 0.5f;
  mz  = (z + zn) * 0.5f;
  ms0 = (s.x + sn.x) * 0.5f;
  ms1 = (s.y + sn.y) * 0.5f;
  ms2 = (s.z + sn.z) * 0.5f;
  ms3 = (s.w + sn.w) * 0.5f;

  float delta = zn - z;
  // softplus(x) = max(x,0) + log1p(exp(-|x|))  (matches jax.nn.softplus)
  float spd = fmaxf(md, 0.0f) + log1pf(expf(-fabsf(md)));
  float a = 1.0f - expf(-spd * delta);

  // level-dependent opacity factor (uses PRE-renorm semantics mids)
  float factor = 1.0f;
  if (lv == 1)      factor = 1.0f - ms0;
  else if (lv == 2) factor = 1.0f - ms0 - ms3;
  a *= factor;

  // level-dependent semantics renormalization
  if (lv == 1) {
    float t = ms1 + ms2 + ms3 + EPS_F;
    ms1 = (ms1 + EPS_F) / t;
    ms2 = (ms2 + EPS_F) / t;
    ms3 = (ms3 + EPS_F) / t;
    ms0 = 0.0f;
  } else if (lv == 2) {
    float t = ms1 + ms2 + EPS_F;
    ms1 = (ms1 + EPS_F) / t;
    ms2 = (ms2 + EPS_F) / t;
    ms0 = 0.0f;
    ms3 = 0.0f;
  }
  alpha = active ? a : 0.0f;
}

// next-sample value for chunk A (lane 31 needs sample 32 = chunk B lane 0)
__device__ __forceinline__ float next_a(float qA, float qB, int lane) {
  float t = __shfl_down(qA, 1);
  float f = __shfl(qB, 0);
  return (lane == 31) ? f : t;
}

__global__ __launch_bounds__(256)
void k_raymarch(const float* __restrict__ colors,
                const float* __restrict__ densities,
                const float* __restrict__ depths,
                const float* __restrict__ sems,
                const int*   __restrict__ levels,
                const int*   __restrict__ ws,
                float*       __restrict__ out)
{
  // 13,824 B of LDS: the whole block's input tile
  __shared__ __align__(16) float  lds_col[RPB * S_ * 3];  // 4608 B
  __shared__ __align__(16) float  lds_den[RPB * S_];      // 1536 B
  __shared__ __align__(16) float  lds_dep[RPB * S_];      // 1536 B
  __shared__ __align__(16) float4 lds_sem[RPB * S_];      // 6144 B

  const int tid  = threadIdx.x;
  const int lane = tid & 31;
  const int wave = tid >> 5;
  const int ray0 = blockIdx.x * RPB;
  const int ray  = ray0 + wave;            // uniform per wave
  const int b    = ray / R_;
  const int lv   = levels[b];              // uniform per wave

  // ---- stage block inputs into LDS with coalesced async b128 copies ----
  const char* gcol = (const char*)(colors    + (size_t)ray0 * S_ * 3);
  const char* gden = (const char*)(densities + (size_t)ray0 * S_);
  const char* gdep = (const char*)(depths    + (size_t)ray0 * S_);
  const char* gsem = (const char*)(sems      + (size_t)ray0 * S_ * 4);

  constexpr int NV_COL = RPB * S_ * 3 / 4;  // 288 float4
  constexpr int NV_SCL = RPB * S_ / 4;      // 96 float4
  constexpr int NV_SEM = RPB * S_;          // 384 float4

  for (int i = tid; i < NV_COL; i += 256)
    async_copy_b128(lds_addr32((const char*)lds_col + i * 16), gcol + i * 16);
  for (int i = tid; i < NV_SCL; i += 256)
    async_copy_b128(lds_addr32((const char*)lds_den + i * 16), gden + i * 16);
  for (int i = tid; i < NV_SCL; i += 256)
    async_copy_b128(lds_addr32((const char*)lds_dep + i * 16), gdep + i * 16);
  for (int i = tid; i < NV_SEM; i += 256)
    async_copy_b128(lds_addr32((const char*)(lds_sem + i)), gsem + i * 16);

  wait_asynccnt0();   // wait this wave's async copies
  __syncthreads();    // then all waves' copies are complete

  const float*  cL = lds_col + wave * (S_ * 3);
  const float*  dL = lds_den + wave * S_;
  const float*  zL = lds_dep + wave * S_;
  const float4* sL = lds_sem + wave * S_;

  // ---- per-lane sample values from LDS ----
  float c0A = cL[lane * 3 + 0];
  float c1A = cL[lane * 3 + 1];
  float c2A = cL[lane * 3 + 2];
  float dA  = dL[lane];
  float zA  = zL[lane];
  float4 sA = sL[lane];

  float c0B = 0.f, c1B = 0.f, c2B = 0.f, dB = 0.f, zB = 0.f;
  float4 sB = make_float4(0.f, 0.f, 0.f, 0.f);
  if (lane < 16) {
    int s = 32 + lane;
    c0B = cL[s * 3 + 0];
    c1B = cL[s * 3 + 1];
    c2B = cL[s * 3 + 2];
    dB  = dL[s];
    zB  = zL[s];
    sB  = sL[s];
  }

  // ---- next-sample values via lane shuffles ----
  float c0An = next_a(c0A, c0B, lane);
  float c1An = next_a(c1A, c1B, lane);
  float c2An = next_a(c2A, c2B, lane);
  float dAn  = next_a(dA,  dB,  lane);
  float zAn  = next_a(zA,  zB,  lane);
  float4 sAn;
  sAn.x = next_a(sA.x, sB.x, lane);
  sAn.y = next_a(sA.y, sB.y, lane);
  sAn.z = next_a(sA.z, sB.z, lane);
  sAn.w = next_a(sA.w, sB.w, lane);

  float c0Bn = __shfl_down(c0B, 1);
  float c1Bn = __shfl_down(c1B, 1);
  float c2Bn = __shfl_down(c2B, 1);
  float dBn  = __shfl_down(dB, 1);
  float zBn  = __shfl_down(zB, 1);
  float4 sBn;
  sBn.x = __shfl_down(sB.x, 1);
  sBn.y = __shfl_down(sB.y, 1);
  sBn.z = __shfl_down(sB.z, 1);
  sBn.w = __shfl_down(sB.w, 1);

  // ---- per-interval midpoint / alpha / semantics ----
  float mc0A, mc1A, mc2A, mzA, ms0A, ms1A, ms2A, ms3A, aA;
  compute_interval(lv, c0A, c0An, c1A, c1An, c2A, c2An, dA, dAn, zA, zAn,
                   sA, sAn, /*active=*/true,
                   mc0A, mc1A, mc2A, mzA, ms0A, ms1A, ms2A, ms3A, aA);

  float mc0B, mc1B, mc2B, mzB, ms0B, ms1B, ms2B, ms3B, aB;
  compute_interval(lv, c0B, c0Bn, c1B, c1Bn, c2B, c2Bn, dB, dBn, zB, zBn,
                   sB, sBn, /*active=*/(lane < 15),
                   mc0B, mc1B, mc2B, mzB, ms0B, ms1B, ms2B, ms3B, aB);

  // ---- transmittance: exclusive prefix product of (1 - alpha + 1e-10) ----
  float tA = 1.0f - aA + 1e-10f;                         // intervals 0..31
  float tB = (lane < 15) ? (1.0f - aB + 1e-10f) : 1.0f;  // intervals 32..46

  float pA = tA;
  for (int off = 1; off < 32; off <<= 1) {
    float v = __shfl_up(pA, off);
    if (lane >= off) pA *= v;
  }
  float eA   = __shfl_up(pA, 1);
  float TA   = (lane == 0) ? 1.0f : eA;
  float totA = __shfl(pA, 31);

  float pB = tB;
  for (int off = 1; off < 32; off <<= 1) {
    float v = __shfl_up(pB, off);
    if (lane >= off) pB *= v;
  }
  float eB = __shfl_up(pB, 1);
  float TB = ((lane == 0) ? 1.0f : eB) * totA;

  float wA = aA * TA;  // intervals 0..31 (all lanes)
  float wB = aB * TB;  // intervals 32..46 (lanes 0..14; else 0)

  // ---- composite sums: 8 channels, wave32 butterfly reduction ----
  float a0 = wA * mc0A + wB * mc0B;
  float a1 = wA * mc1A + wB * mc1B;
  float a2 = wA * mc2A + wB * mc2B;
  float az = wA * mzA  + wB * mzB;
  float q0 = wA * ms0A + wB * ms0B;
  float q1 = wA * ms1A + wB * ms1B;
  float q2 = wA * ms2A + wB * ms2B;
  float q3 = wA * ms3A + wB * ms3B;
  for (int off = 16; off; off >>= 1) {
    a0 += __shfl_xor(a0, off);
    a1 += __shfl_xor(a1, off);
    a2 += __shfl_xor(a2, off);
    az += __shfl_xor(az, off);
    q0 += __shfl_xor(q0, off);
    q1 += __shfl_xor(q1, off);
    q2 += __shfl_xor(q2, off);
    q3 += __shfl_xor(q3, off);
  }

  // ---- outputs (concatenated flat: rgb | depth | sem | weights) ----
  float* out_rgb   = out;
  float* out_depth = out + (size_t)NRAYS * 3;
  float* out_sem   = out + (size_t)NRAYS * 4;
  float* out_w     = out + (size_t)NRAYS * 8;

  if (lane == 0) {
    out_rgb[(size_t)ray * 3 + 0] = a0;
    out_rgb[(size_t)ray * 3 + 1] = a1;
    out_rgb[(size_t)ray * 3 + 2] = a2;

    float dmin = __int_as_float(ws[0]);
    float dmax = __int_as_float(ws[1]);
    float cd = az;
    if (cd != cd) cd = __int_as_float(0x7F800000);  // NaN -> +inf
    cd = fminf(fmaxf(cd, dmin), dmax);
    out_depth[ray] = cd;

    out_sem[(size_t)ray * 4 + 0] = q0;
    out_sem[(size_t)ray * 4 + 1] = q1;
    out_sem[(size_t)ray * 4 + 2] = q2;
    out_sem[(size_t)ray * 4 + 3] = q3;
  }

  // weights: coalesced per-interval stores
  out_w[(size_t)ray * NM1 + lane] = wA;
  if (lane < 15) out_w[(size_t)ray * NM1 + 32 + lane] = wB;
}

// ---------------------------------------------------------------------------
// Host launcher
// ---------------------------------------------------------------------------
extern "C" void kernel_launch(void* const* d_in, const int* in_sizes, int n_in,
                              void* d_out, int out_size, void* d_ws, size_t ws_size,
                              hipStream_t stream) {
  (void)in_sizes; (void)n_in; (void)out_size; (void)ws_size;
  const float* colors    = (const float*)d_in[0];
  const float* densities = (const float*)d_in[1];
  const float* depths    = (const float*)d_in[2];
  const float* semantics = (const float*)d_in[3];
  const int*   levels    = (const int*)d_in[4];
  float* out = (float*)d_out;
  int*   ws  = (int*)d_ws;

  k_init_minmax<<<1, 1, 0, stream>>>(ws);
  k_depth_minmax<<<1024, 256, 0, stream>>>(depths, ws, NRAYS * S_);
  k_raymarch<<<NRAYS / RPB, 256, 0, stream>>>(colors, densities, depths, semantics,
                                              levels, ws, out);
}